// RelationConv_32985348833527
// MI455X (gfx1250) — compile-verified
//
#include <hip/hip_runtime.h>
#include <stdint.h>

// ---------------- CDNA5 (gfx1250, wave32) tuning ----------------
#define WAVE        32
#define WPB         8                 // waves (=nodes) per block
#define NODE_BLOCK  (WAVE * WPB)      // 256 threads
#define CHUNK       128               // staged edges per wave per chunk (8B pairs -> 1KB/wave)
#define SCAN_T      1024

// K0: zero accumulators
__global__ void k0_init(float* __restrict__ sq, int* __restrict__ deg, int N) {
    int i = blockIdx.x * blockDim.x + threadIdx.x;
    if (i < N) { sq[i] = 0.0f; deg[i] = 0; }
}

// K1: per-source sum of squares of edge_attr + degree counts (L2 atomics)
__global__ void k1_edge_norm(const int* __restrict__ row, const float* __restrict__ ea,
                             float* __restrict__ sq, int* __restrict__ deg, int E) {
    int e = blockIdx.x * blockDim.x + threadIdx.x;
    if (e >= E) return;
    int r = row[e];
    float a = ea[e];
    unsafeAtomicAdd(&sq[r], a * a);   // native global_atomic_add_f32
    atomicAdd(&deg[r], 1);
}

// K2: one wave32 per node: xn = x / max(||x||,1e-12); finalize srcnorm in-place in sq
__global__ __launch_bounds__(NODE_BLOCK) void k2_node_norm(const float* __restrict__ x,
                                                           float* __restrict__ xn,
                                                           float* __restrict__ sq, int N) {
    int wv = threadIdx.x >> 5;
    int lane = threadIdx.x & 31;
    int node = blockIdx.x * WPB + wv;
    if (node >= N) return;
    const float2* x2 = (const float2*)x + (size_t)node * 32;
    float2 v = x2[lane];
    float ss = v.x * v.x + v.y * v.y;
#pragma unroll
    for (int off = 16; off > 0; off >>= 1) ss += __shfl_xor(ss, off, 32);
    float inv = 1.0f / fmaxf(sqrtf(ss), 1e-12f);
    float2 o; o.x = v.x * inv; o.y = v.y * inv;
    ((float2*)xn + (size_t)node * 32)[lane] = o;
    if (lane == 0) sq[node] = fmaxf(sqrtf(sq[node]), 1e-12f);  // srcnorm
}

// K3: single-block exclusive scan deg -> rowptr (+cursor copy)
__global__ void k3_scan(const int* __restrict__ deg, int* __restrict__ rowptr,
                        int* __restrict__ cursor, int N, int E) {
    __shared__ int buf[SCAN_T];
    __shared__ int carry;
    int tid = threadIdx.x;
    if (tid == 0) carry = 0;
    __syncthreads();
    for (int base = 0; base < N; base += SCAN_T) {
        int i = base + tid;
        int v = (i < N) ? deg[i] : 0;
        buf[tid] = v;
        __syncthreads();
        for (int off = 1; off < SCAN_T; off <<= 1) {
            int t = (tid >= off) ? buf[tid - off] : 0;
            __syncthreads();
            buf[tid] += t;
            __syncthreads();
        }
        int c = carry;
        int excl = c + buf[tid] - v;
        if (i < N) { rowptr[i] = excl; cursor[i] = excl; }
        int total = buf[SCAN_T - 1];
        __syncthreads();
        if (tid == 0) carry = c + total;
        __syncthreads();
    }
    if (tid == 0) rowptr[N] = E;
}

// K4: scatter edges into CSR order as packed {col, ea_normalized} 8B pairs
__global__ void k4_scatter(const int* __restrict__ row, const int* __restrict__ col,
                           const float* __restrict__ ea, const float* __restrict__ snorm,
                           int* __restrict__ cursor, uint64_t* __restrict__ pairs, int E) {
    int e = blockIdx.x * blockDim.x + threadIdx.x;
    if (e >= E) return;
    int r = row[e];
    float a = ea[e] / snorm[r];
    int pos = atomicAdd(&cursor[r], 1);
    pairs[pos] = ((uint64_t)__float_as_uint(a) << 32) | (uint32_t)col[e];
}

// K5: one wave32 per node. Async-stage the node's CSR edge segment into LDS
// (gfx1250 GLOBAL_LOAD_ASYNC_TO_LDS + S_WAIT_ASYNCCNT), online softmax, gather xn[col].
__global__ __launch_bounds__(NODE_BLOCK) void k5_gather(const float* __restrict__ xn,
                                                        const uint64_t* __restrict__ pairs,
                                                        const int* __restrict__ rowptr,
                                                        const float* __restrict__ epsP,
                                                        const float* __restrict__ betaP,
                                                        float* __restrict__ out, int N) {
    __shared__ uint64_t lds[WPB * CHUNK];
    int wv = threadIdx.x >> 5;
    int lane = threadIdx.x & 31;
    int node = blockIdx.x * WPB + wv;
    if (node >= N) return;

    float beta = betaP[0];
    int start = rowptr[node];
    int end   = rowptr[node + 1];

    // online softmax state; self loop (alpha = beta*1) is the initial element
    float  m = beta;            // running max
    float  s = 1.0f;            // running sum = exp(beta - m) so far
    float2 acc = make_float2(0.0f, 0.0f);

    uint64_t* slot = &lds[wv * CHUNK];
    unsigned ldsBase = (unsigned)(uintptr_t)slot;   // low 32 bits of flat LDS addr = LDS offset

    for (int cb = start; cb < end; cb += CHUNK) {
        int rem = end - cb;
        int cnt = rem < CHUNK ? rem : CHUNK;
        // Stage CHUNK pairs (uniform issue count; pairs buffer is padded by CHUNK)
#pragma unroll
        for (int t = 0; t < CHUNK / WAVE; ++t) {
            int e = t * WAVE + lane;
            unsigned la = ldsBase + (unsigned)e * 8u;
            uint64_t ga = (uint64_t)(uintptr_t)(pairs + cb + e);
            asm volatile("global_load_async_to_lds_b64 %0, %1, off"
                         :: "v"(la), "v"(ga) : "memory");
        }
        asm volatile("s_wait_asynccnt 0x0" ::: "memory");

        // chunk max (uniform LDS broadcast reads)
        float cm = m;
        for (int e = 0; e < cnt; ++e) {
            float a = __uint_as_float((uint32_t)(slot[e] >> 32));
            cm = fmaxf(cm, beta * a);
        }
        float scale = __expf(m - cm);
        s *= scale; acc.x *= scale; acc.y *= scale; m = cm;

        // accumulate exp-weights and gathered features (coalesced float2 per edge)
        for (int e = 0; e < cnt; ++e) {
            uint64_t p = slot[e];
            float a = __uint_as_float((uint32_t)(p >> 32));
            uint32_t c = (uint32_t)p;
            float w = __expf(beta * a - m);
            s += w;
            float2 v = ((const float2*)xn + (size_t)c * 32)[lane];
            acc.x += w * v.x;
            acc.y += w * v.y;
        }
    }

    float inv  = 1.0f / s;
    float coef = (1.0f + epsP[0]) + __expf(beta - m) * inv;   // (1+eps) + self-loop alpha
    float2 vi = ((const float2*)xn + (size_t)node * 32)[lane];
    float2 o;
    o.x = coef * vi.x + acc.x * inv;
    o.y = coef * vi.y + acc.y * inv;
    ((float2*)out + (size_t)node * 32)[lane] = o;
}

extern "C" void kernel_launch(void* const* d_in, const int* in_sizes, int n_in,
                              void* d_out, int out_size, void* d_ws, size_t ws_size,
                              hipStream_t stream) {
    const float* x    = (const float*)d_in[0];
    const int*   ei   = (const int*)d_in[1];
    const float* ea   = (const float*)d_in[2];
    const float* eps  = (const float*)d_in[3];
    const float* beta = (const float*)d_in[4];
    const int N = in_sizes[0] / 64;
    const int E = in_sizes[2];
    const int* row = ei;          // edge_index[0]
    const int* col = ei + E;      // edge_index[1]

    auto alignup = [](size_t v) { return (v + 255) & ~(size_t)255; };
    char* ws = (char*)d_ws;
    size_t o = 0;
    float*    sq     = (float*)(ws + o);    o += alignup((size_t)N * 4);        // sumsq -> srcnorm
    int*      deg    = (int*)(ws + o);      o += alignup((size_t)N * 4);
    int*      rowptr = (int*)(ws + o);      o += alignup((size_t)(N + 1) * 4);
    int*      cursor = (int*)(ws + o);      o += alignup((size_t)N * 4);
    float*    xn     = (float*)(ws + o);    o += alignup((size_t)N * 64 * 4);
    uint64_t* pairs  = (uint64_t*)(ws + o); o += alignup((size_t)(E + CHUNK) * 8);
    (void)ws_size; (void)n_in; (void)out_size;

    float* out = (float*)d_out;
    const int tb = 256;
    const int nodeBlocks = (N + WPB - 1) / WPB;

    k0_init     <<<(N + tb - 1) / tb, tb, 0, stream>>>(sq, deg, N);
    k1_edge_norm<<<(E + tb - 1) / tb, tb, 0, stream>>>(row, ea, sq, deg, E);
    k2_node_norm<<<nodeBlocks, NODE_BLOCK, 0, stream>>>(x, xn, sq, N);
    k3_scan     <<<1, SCAN_T, 0, stream>>>(deg, rowptr, cursor, N, E);
    k4_scatter  <<<(E + tb - 1) / tb, tb, 0, stream>>>(row, col, ea, sq, cursor, pairs, E);
    k5_gather   <<<nodeBlocks, NODE_BLOCK, 0, stream>>>(xn, pairs, rowptr, eps, beta, out, N);
}